// DocREModel_9174050144906
// MI455X (gfx1250) — compile-verified
//
#include <hip/hip_runtime.h>
#include <hip/hip_bf16.h>

// ---------------------------------------------------------------------------
// Problem constants (from the reference)
// ---------------------------------------------------------------------------
#define NN    512          // nodes
#define DD    768          // hidden
#define FH0   64           // layer-0 per-head dim
#define HEADS 12
#define NPAIR (NN * NN)    // 262144
#define VCOLS 32           // 24 used (12 heads x 2 layers), padded to 32
#define ALPHA 0.2f
#define NEGV  -9e15f

typedef __attribute__((ext_vector_type(2))) float v2f;
typedef __attribute__((ext_vector_type(8))) float v8f;

// ---------------------------------------------------------------------------
// Generic batched fp32 WMMA GEMM:  C[b] = A[b] (MxK, lda) * B[b] (KxN, ldb)
// Row-major. One wave computes a 16x32 strip of C (two 16x16 WMMA tiles that
// share the A fragment -> A loaded once per K-step). K unrolled by 4 so 20
// loads are issued as a batch followed by 8 back-to-back
// V_WMMA_F32_16X16X4_F32 ops, letting the scheduler hide load latency.
//
// Requirements: M % 16 == 0, N % 32 == 0, K % 16 == 0 (true for all calls).
//
// Fragment layouts (ISA 7.12.2, wave32):
//   A 16x4:  lane l holds row M=l%16; VGPR0 = K=2*(l/16), VGPR1 = K=2*(l/16)+1
//   B 4x16:  lane l holds col N=l%16; VGPR0 = K=2*(l/16), VGPR1 = K=2*(l/16)+1
//   C 16x16: lane l, VGPR r -> M = r + 8*(l/16), N = l%16
// ---------------------------------------------------------------------------
__global__ void wmma_gemm_f32(const float* __restrict__ A,
                              const float* __restrict__ B,
                              float* __restrict__ C,
                              int M, int N, int K,
                              long sA, long sB, long sC,
                              int lda, int ldb, int ldc)
{
    const int tileN = blockIdx.x;           // N/32 strips
    const int tileM = blockIdx.y;           // M/16
    const int b     = blockIdx.z;
    const int lane  = threadIdx.x;          // 0..31 (wave32)

    const float* Ab = A + (long)b * sA;
    const float* Bb = B + (long)b * sB;
    float*       Cb = C + (long)b * sC;

    const int m     = tileM * 16 + (lane & 15);
    const int n0    = tileN * 32 + (lane & 15);
    const int khalf = (lane >> 4) * 2;      // 0 or 2

    const float* arow  = Ab + (long)m * lda + khalf;       // A row, khalf-offset
    const float* bcol0 = Bb + (long)khalf * ldb + n0;      // B col strip 0
    const float* bcol1 = bcol0 + 16;                       // B col strip 1

    v8f acc0 = {}, acc1 = {};
    for (int k0 = 0; k0 < K; k0 += 16) {
        v2f av[4], b0[4], b1[4];
#pragma unroll
        for (int u = 0; u < 4; ++u) {
            const int k = k0 + 4 * u;
            av[u] = *(const v2f*)(arow + k);               // global_load_b64
            b0[u].x = bcol0[(long)k * ldb];
            b0[u].y = bcol0[(long)(k + 1) * ldb];
            b1[u].x = bcol1[(long)k * ldb];
            b1[u].y = bcol1[(long)(k + 1) * ldb];
        }
        // speculative stream-ahead on the A row (dominant e-tensor stream)
        __builtin_prefetch(arow + k0 + 64, 0, 3);
#pragma unroll
        for (int u = 0; u < 4; ++u) {
            acc0 = __builtin_amdgcn_wmma_f32_16x16x4_f32(
                       false, av[u], false, b0[u], (short)0, acc0, false, false);
            acc1 = __builtin_amdgcn_wmma_f32_16x16x4_f32(
                       false, av[u], false, b1[u], (short)0, acc1, false, false);
        }
    }

    const int rowBase = tileM * 16 + (lane >> 4) * 8;
    const int col0    = tileN * 32 + (lane & 15);
#pragma unroll
    for (int r = 0; r < 8; ++r) {
        Cb[(long)(rowBase + r) * ldc + col0]      = acc0[r];
        Cb[(long)(rowBase + r) * ldc + col0 + 16] = acc1[r];
    }
}

// ---------------------------------------------------------------------------
// t[l,h,i] = sum_f W[h,i,f] * a3[h,f]      (tiny GEMV)
// ---------------------------------------------------------------------------
__global__ void k_tvec(const float* __restrict__ W0, const float* __restrict__ a0,
                       const float* __restrict__ W1, const float* __restrict__ a1,
                       float* __restrict__ t)
{
    int tid = blockIdx.x * blockDim.x + threadIdx.x;
    if (tid >= 2 * HEADS * DD) return;
    int layer = tid / (HEADS * DD);
    int rem   = tid % (HEADS * DD);
    int h = rem / DD, i = rem % DD;
    float acc = 0.f;
    if (layer == 0) {
        const float* w = W0 + ((long)h * DD + i) * FH0;
        const float* a = a0 + h * (3 * FH0) + 2 * FH0;
        for (int f = 0; f < FH0; ++f) acc += w[f] * a[f];
    } else {
        const float* w = W1 + ((long)h * DD + i) * DD;
        const float* a = a1 + h * (3 * DD) + 2 * DD;
        for (int f = 0; f < DD; ++f) acc += w[f] * a[f];
    }
    t[tid] = acc;
}

// ---------------------------------------------------------------------------
// Vt[k, col] = sum_i We[h, k, i] * t[l, h, i]     col = l*12 + h  (24 cols),
// cols 24..31 zero-padded.  Vt is the row-major [768 x 32] B operand for the
// single fused e-stream GEMM covering both layers.
// ---------------------------------------------------------------------------
__global__ void k_vt(const float* __restrict__ We0, const float* __restrict__ We1,
                     const float* __restrict__ t, float* __restrict__ Vt)
{
    int tid = blockIdx.x * blockDim.x + threadIdx.x;
    if (tid >= DD * VCOLS) return;
    int k = tid / VCOLS, col = tid % VCOLS;
    float acc = 0.f;
    if (col < 2 * HEADS) {
        int layer = col / HEADS, h = col % HEADS;
        const float* We = (layer == 0) ? We0 : We1;
        const float* w  = We + ((long)h * DD + k) * DD;
        const float* tv = t + (long)layer * HEADS * DD + (long)h * DD;
        for (int i = 0; i < DD; ++i) acc += w[i] * tv[i];
    }
    Vt[tid] = acc;
}

// ---------------------------------------------------------------------------
// s_src[h,n] = h[h,n,:].a1 ; s_dst[h,n] = h[h,n,:].a2
// ---------------------------------------------------------------------------
__global__ void k_srcdst(const float* __restrict__ hmat, const float* __restrict__ a,
                         int F, int aStride,
                         float* __restrict__ ssrc, float* __restrict__ sdst)
{
    int tid = blockIdx.x * blockDim.x + threadIdx.x;
    if (tid >= HEADS * NN) return;
    int h = tid / NN, n = tid % NN;
    const float* hv = hmat + ((long)h * NN + n) * F;
    const float* p1 = a + (long)h * aStride;
    const float* p2 = p1 + F;
    float s1 = 0.f, s2 = 0.f;
    for (int f = 0; f < F; ++f) { s1 += hv[f] * p1[f]; s2 += hv[f] * p2[f]; }
    ssrc[tid] = s1;
    sdst[tid] = s2;
}

// ---------------------------------------------------------------------------
// Masked leaky-relu softmax over row j. sea layout: sea[(i*NN+j)*32 + heOff+h]
// ---------------------------------------------------------------------------
__global__ void k_attn_softmax(const float* __restrict__ ssrc,
                               const float* __restrict__ sdst,
                               const float* __restrict__ sea,
                               const int*   __restrict__ adj,
                               float* __restrict__ att, int heOff)
{
    const int i = blockIdx.x;
    const int h = blockIdx.y;
    const int t = threadIdx.x;              // 0..255
    __shared__ float srow[NN];
    __shared__ float red[256];

    const float si = ssrc[h * NN + i];
    float lmax = -3.4e38f;
    for (int j = t; j < NN; j += 256) {
        float s = si + sdst[h * NN + j]
                + sea[((long)i * NN + j) * VCOLS + heOff + h];
        s = (s > 0.f) ? s : ALPHA * s;                 // leaky relu
        if (adj[i * NN + j] <= 0) s = NEGV;            // mask
        srow[j] = s;
        lmax = fmaxf(lmax, s);
    }
    red[t] = lmax; __syncthreads();
    for (int o = 128; o > 0; o >>= 1) {
        if (t < o) red[t] = fmaxf(red[t], red[t + o]);
        __syncthreads();
    }
    const float m = red[0]; __syncthreads();

    float lsum = 0.f;
    for (int j = t; j < NN; j += 256) {
        float ex = __expf(srow[j] - m);
        srow[j] = ex;
        lsum += ex;
    }
    red[t] = lsum; __syncthreads();
    for (int o = 128; o > 0; o >>= 1) {
        if (t < o) red[t] += red[t + o];
        __syncthreads();
    }
    const float inv = 1.f / red[0];
    for (int j = t; j < NN; j += 256)
        att[((long)h * NN + i) * NN + j] = srow[j] * inv;
}

// ---------------------------------------------------------------------------
// x1[n, h*64+f] = elu(hp0[h,n,f])     (layer-0 concat epilogue)
// ---------------------------------------------------------------------------
__global__ void k_x1(const float* __restrict__ hp0, float* __restrict__ x1)
{
    int tid = blockIdx.x * blockDim.x + threadIdx.x;
    if (tid >= HEADS * NN * FH0) return;
    int h = tid / (NN * FH0);
    int rem = tid % (NN * FH0);
    int n = rem / FH0, f = rem % FH0;
    float v = hp0[tid];
    v = (v > 0.f) ? v : (__expf(v) - 1.f);
    x1[(long)n * DD + h * FH0 + f] = v;
}

// ---------------------------------------------------------------------------
// out[n,f] = elu(mean_h hp1[h,n,f])    (layer-1 islast epilogue)
// ---------------------------------------------------------------------------
__global__ void k_out(const float* __restrict__ hp1, float* __restrict__ out)
{
    int tid = blockIdx.x * blockDim.x + threadIdx.x;
    if (tid >= NN * DD) return;
    int n = tid / DD, f = tid % DD;
    float s = 0.f;
    for (int h = 0; h < HEADS; ++h)
        s += hp1[((long)h * NN + n) * DD + f];
    s *= (1.f / 12.f);
    out[tid] = (s > 0.f) ? s : (__expf(s) - 1.f);
}

// ---------------------------------------------------------------------------
extern "C" void kernel_launch(void* const* d_in, const int* in_sizes, int n_in,
                              void* d_out, int out_size, void* d_ws, size_t ws_size,
                              hipStream_t stream)
{
    (void)in_sizes; (void)n_in; (void)out_size; (void)ws_size;

    const float* x   = (const float*)d_in[0];
    const int*   adj = (const int*)  d_in[1];
    const float* e   = (const float*)d_in[2];
    const float* We0 = (const float*)d_in[3];
    const float* W0  = (const float*)d_in[4];
    const float* a0  = (const float*)d_in[5];
    const float* We1 = (const float*)d_in[6];
    const float* W1  = (const float*)d_in[7];
    const float* a1  = (const float*)d_in[8];
    float* out = (float*)d_out;

    // ---- workspace layout (floats), 256B aligned bump allocator ----
    char* ws = (char*)d_ws;
    size_t off = 0;
    auto alloc = [&](size_t nflt) {
        float* p = (float*)(ws + off);
        off += ((nflt * sizeof(float)) + 255) & ~size_t(255);
        return p;
    };
    float* t_buf = alloc(2 * HEADS * DD);             //  73.7 KB
    float* Vt    = alloc(DD * VCOLS);                 //  98  KB
    float* h0    = alloc((size_t)HEADS * NN * FH0);   // 1.57 MB
    float* ssrc0 = alloc(HEADS * NN);
    float* sdst0 = alloc(HEADS * NN);
    float* ssrc1 = alloc(HEADS * NN);
    float* sdst1 = alloc(HEADS * NN);
    float* hp0   = alloc((size_t)HEADS * NN * FH0);   // 1.57 MB
    float* x1    = alloc((size_t)NN * DD);            // 1.57 MB
    float* h1    = alloc((size_t)HEADS * NN * DD);    // 18.9 MB
    float* att   = alloc((size_t)HEADS * NN * NN);    // 12.6 MB (reused layer0+1)
    float* sea   = alloc((size_t)NPAIR * VCOLS);      // 33.6 MB
    float* hp1   = sea;  // alias: sea dead after layer-1 softmax; hp1 = 18.9 MB

    const int TB = 256;
    auto blks = [](long n, int tb) { return (unsigned)((n + tb - 1) / tb); };

    // 1) t = W . a3 (both layers)
    k_tvec<<<blks(2L * HEADS * DD, TB), TB, 0, stream>>>(W0, a0, W1, a1, t_buf);

    // 2) Vt[768 x 32] = We^T . t, transposed + zero-padded for GEMM B operand
    k_vt<<<blks((long)DD * VCOLS, TB), TB, 0, stream>>>(We0, We1, t_buf, Vt);

    // 3) Fused e-stream GEMM: sea[262144 x 32] = E[262144 x 768] * Vt[768 x 32]
    //    -> single read of the 805 MB e tensor covers BOTH layers.
    {
        dim3 grid(VCOLS / 32, NPAIR / 16, 1);
        wmma_gemm_f32<<<grid, 32, 0, stream>>>(e, Vt, sea,
                                               NPAIR, VCOLS, DD,
                                               0, 0, 0,
                                               DD, VCOLS, VCOLS);
    }

    // 4) h0[h] = x * W0[h]   (M=512, N=64, K=768, batch=12)
    {
        dim3 grid(FH0 / 32, NN / 16, HEADS);
        wmma_gemm_f32<<<grid, 32, 0, stream>>>(x, W0, h0,
                                               NN, FH0, DD,
                                               0, (long)DD * FH0, (long)NN * FH0,
                                               DD, FH0, FH0);
    }

    // 5) layer-0 src/dst scores
    k_srcdst<<<blks((long)HEADS * NN, TB), TB, 0, stream>>>(h0, a0, FH0, 3 * FH0,
                                                            ssrc0, sdst0);

    // 6) layer-0 masked softmax  (heOff = 0..11 in sea columns)
    {
        dim3 grid(NN, HEADS);
        k_attn_softmax<<<grid, 256, 0, stream>>>(ssrc0, sdst0, sea, adj, att, 0);
    }

    // 7) hp0[h] = att[h] * h0[h]   (M=512, N=64, K=512)
    {
        dim3 grid(FH0 / 32, NN / 16, HEADS);
        wmma_gemm_f32<<<grid, 32, 0, stream>>>(att, h0, hp0,
                                               NN, FH0, NN,
                                               (long)NN * NN, (long)NN * FH0, (long)NN * FH0,
                                               NN, FH0, FH0);
    }

    // 8) x1 = elu + head-concat
    k_x1<<<blks((long)HEADS * NN * FH0, TB), TB, 0, stream>>>(hp0, x1);

    // 9) h1[h] = x1 * W1[h]   (M=512, N=768, K=768, batch=12)
    {
        dim3 grid(DD / 32, NN / 16, HEADS);
        wmma_gemm_f32<<<grid, 32, 0, stream>>>(x1, W1, h1,
                                               NN, DD, DD,
                                               0, (long)DD * DD, (long)NN * DD,
                                               DD, DD, DD);
    }

    // 10) layer-1 src/dst scores
    k_srcdst<<<blks((long)HEADS * NN, TB), TB, 0, stream>>>(h1, a1, DD, 3 * DD,
                                                            ssrc1, sdst1);

    // 11) layer-1 masked softmax (heOff = 12..23 in sea columns)
    {
        dim3 grid(NN, HEADS);
        k_attn_softmax<<<grid, 256, 0, stream>>>(ssrc1, sdst1, sea, adj, att, HEADS);
    }

    // 12) hp1[h] = att[h] * h1[h]   (M=512, N=768, K=512) ; hp1 aliases sea
    {
        dim3 grid(DD / 32, NN / 16, HEADS);
        wmma_gemm_f32<<<grid, 32, 0, stream>>>(att, h1, hp1,
                                               NN, DD, NN,
                                               (long)NN * NN, (long)NN * DD, (long)NN * DD,
                                               NN, DD, DD);
    }

    // 13) mean over heads + elu -> out
    k_out<<<blks((long)NN * DD, TB), TB, 0, stream>>>(hp1, out);
}